// BasisExpansionLayer_13821204759158
// MI455X (gfx1250) — compile-verified
//
#include <hip/hip_runtime.h>
#include <hip/hip_bf16.h>
#include <math.h>

// BasisExpansionLayer: out = concat([sin(x), (x_i * x_j).reshape(B, D*D), x], axis=1)
// B=256, D=512. Output row length = D + D*D + D = 263168 f32.
// Memory-bound: ~269.5 MB of stores -> ~11.6 us floor @ 23.3 TB/s.
// Pair part computed as 16x16 rank-1 outer-product tiles via
// V_WMMA_F32_16X16X4_F32 (K=0 carries data, K=1..3 zeroed -> exact f32).

typedef __attribute__((ext_vector_type(2))) float v2f;
typedef __attribute__((ext_vector_type(8))) float v8f;

#define BATCH      256
#define DIM        512
#define OUT_ROW    (DIM + DIM * DIM + DIM)   // 263168
#define PAIR_OFF   DIM
#define IDENT_OFF  (DIM + DIM * DIM)

// ---------------------------------------------------------------------------
// Kernel 1: sin(x) and identity slices (1 MB total, trivial edge work).
// ---------------------------------------------------------------------------
__global__ void __launch_bounds__(256)
basis_sin_ident_kernel(const float* __restrict__ x, float* __restrict__ out) {
    const int b = blockIdx.x;
    const float* __restrict__ xr = x + (size_t)b * DIM;
    float* __restrict__ orow = out + (size_t)b * OUT_ROW;
    for (int d = threadIdx.x; d < DIM; d += blockDim.x) {
        float v = xr[d];
        __builtin_nontemporal_store(sinf(v), orow + d);               // sin slice
        __builtin_nontemporal_store(v, orow + IDENT_OFF + d);         // identity slice
    }
}

// ---------------------------------------------------------------------------
// Kernel 2: pair part. One block per (row-quarter, batch). 8 waves/block,
// each wave owns one 16-row tile stripe (ti) and sweeps 32 tiles in tj.
// Outer product per tile via v_wmma_f32_16x16x4_f32 with K=0 only.
//
// A (16x4 f32) VGPR layout: VGPR0 lanes0-15 = {M=lane, K=0}; all else zeroed.
// B (4x16 f32): K=0 row (N=0..15) striped across VGPR0 lanes0-15; rest zeroed.
// C/D (16x16 f32): VGPR r holds row r (lanes 0-15, N=lane) and row r+8
// (lanes 16-31, N=lane-16).
// ---------------------------------------------------------------------------
__global__ void __launch_bounds__(256)
basis_pair_wmma_kernel(const float* __restrict__ x, float* __restrict__ out) {
    __shared__ float xrow[DIM];

    const int b = blockIdx.y;
    const float* __restrict__ xg = x + (size_t)b * DIM;

    // Stage this batch row into LDS (512 floats = 2 KB).
    for (int i = threadIdx.x; i < DIM; i += blockDim.x)
        xrow[i] = xg[i];
    __syncthreads();

    const int lane = threadIdx.x & 31;
    const int wave = threadIdx.x >> 5;          // 0..7
    const int ti   = blockIdx.x * 8 + wave;     // 0..31 tile-row index
    const int l15  = lane & 15;
    const bool lo  = lane < 16;
    const int mofs = lo ? 0 : 8;                // row offset this lane covers in C/D

    // Build A operand once per wave: x[b, ti*16 + m] at K=0, zeros elsewhere.
    float at = xrow[ti * 16 + l15];
    v2f a;
    a.x = lo ? at : 0.0f;   // lanes 0-15: {M=lane, K=0}; lanes 16-31 ({M,K=2}) = 0
    a.y = 0.0f;             // K=1 / K=3 rows = 0

    float* __restrict__ pbase =
        out + (size_t)b * OUT_ROW + PAIR_OFF + (size_t)(ti * 16) * DIM;

    for (int tj = 0; tj < 32; ++tj) {
        // B operand: x[b, tj*16 + n] at K=0 row, zeros elsewhere.
        float bt = xrow[tj * 16 + l15];
        v2f bm;
        bm.x = lo ? bt : 0.0f;
        bm.y = 0.0f;

        v8f acc = {};
        // 8 args: (neg_a, A, neg_b, B, c_mod, C, reuse_a, reuse_b)
        acc = __builtin_amdgcn_wmma_f32_16x16x4_f32(
            /*neg_a=*/false, a, /*neg_b=*/false, bm,
            /*c_mod=*/(short)0, acc, /*reuse_a=*/false, /*reuse_b=*/false);

        // Lane writes D[mofs+r][l15] for r=0..7:
        // rows ti*16+mofs+r, cols tj*16+l15. NT stores (output >> L2).
        float* __restrict__ cbase = pbase + (size_t)(tj * 16 + l15);
        #pragma unroll
        for (int r = 0; r < 8; ++r) {
            __builtin_nontemporal_store(acc[r], cbase + (size_t)(r + mofs) * DIM);
        }
    }
}

// ---------------------------------------------------------------------------
extern "C" void kernel_launch(void* const* d_in, const int* in_sizes, int n_in,
                              void* d_out, int out_size, void* d_ws, size_t ws_size,
                              hipStream_t stream) {
    (void)in_sizes; (void)n_in; (void)out_size; (void)d_ws; (void)ws_size;
    const float* x = (const float*)d_in[0];
    float* out = (float*)d_out;

    // Edges: sin + identity (disjoint output regions, no inter-kernel dep).
    basis_sin_ident_kernel<<<dim3(BATCH), dim3(256), 0, stream>>>(x, out);

    // Pair part: grid (4 ti-quarters, 256 batch rows), 8 waves/block.
    basis_pair_wmma_kernel<<<dim3(4, BATCH), dim3(256), 0, stream>>>(x, out);
}